// CrossAug_30176440221860
// MI455X (gfx1250) — compile-verified
//
#include <hip/hip_runtime.h>
#include <math.h>

// ---------------- problem constants (from reference) ----------------
#define NS_C   30000
#define D1U_C  100000
#define D2U_C  80000
#define NU_C   150000      // D1U + D2U - NS
#define D1I_C  50000
#define D2I_C  40000
#define N1_C   200000      // NU + D1I
#define N2_C   190000      // NU + D2I
#define EMB_C  128
#define ESTR   384         // 3 * 128 : [x0 | x1 | x2] per row
#define NNZ1_C 4000000
#define NNZ2_C 3800000

typedef float v2f __attribute__((ext_vector_type(2)));
typedef float v8f __attribute__((ext_vector_type(8)));

// ---------------- init: e[:, 0:128] = concat(user, item) ----------------
__global__ void init_x0(const float* __restrict__ user,
                        const float* __restrict__ item,
                        float* __restrict__ e, int nu, int n) {
    int t = blockIdx.x * blockDim.x + threadIdx.x;       // one float4 per thread
    if (t >= n * 32) return;
    int row = t >> 5, j = t & 31;
    const float4* src = (row < nu)
        ? (const float4*)(user + (size_t)row * EMB_C)
        : (const float4*)(item + (size_t)(row - nu) * EMB_C);
    *(float4*)(e + (size_t)row * ESTR + j * 4) = src[j];
}

// ---------------- zero fill (side buffers) ----------------
__global__ void zero_f4(float* __restrict__ p, long n4) {
    long t = blockIdx.x * (long)blockDim.x + threadIdx.x;
    long stride = gridDim.x * (long)blockDim.x;
    float4 z = make_float4(0.f, 0.f, 0.f, 0.f);
    for (long i = t; i < n4; i += stride) ((float4*)p)[i] = z;
}

// ---------------- SpMM: side[rows[k]] += vals[k] * x[cols[k]] ----------------
// One wave per nnz: lane l handles columns 4l..4l+3 (float4 gather + 4 HW f32 atomics).
__global__ void spmm_atomic(const int* __restrict__ rows,
                            const int* __restrict__ cols,
                            const float* __restrict__ vals,
                            const float* __restrict__ e_x,   // row stride ESTR
                            float* __restrict__ side,        // row stride 128
                            int nnz) {
    int  lane = threadIdx.x & 31;
    long wid  = (blockIdx.x * (long)blockDim.x + threadIdx.x) >> 5;
    long nw   = (gridDim.x * (long)blockDim.x) >> 5;
    for (long k = wid; k < nnz; k += nw) {
        if (k + nw < nnz) {                       // prefetch index stream ahead
            __builtin_prefetch(cols + k + nw, 0, 0);
            __builtin_prefetch(vals + k + nw, 0, 0);
        }
        int   r = rows[k];
        int   c = cols[k];
        float v = vals[k];
        float4 xv = *(const float4*)(e_x + (size_t)c * ESTR + lane * 4);
        float* dst = side + (size_t)r * EMB_C + lane * 4;
        unsafeAtomicAdd(dst + 0, v * xv.x);       // global_atomic_add_f32, no-return
        unsafeAtomicAdd(dst + 1, v * xv.y);
        unsafeAtomicAdd(dst + 2, v * xv.z);
        unsafeAtomicAdd(dst + 3, v * xv.w);
    }
}

// ---------------- fused: g = x + s + x*s for both domains, + transfer on user rows ----------------
__global__ void combine_transfer(const float* __restrict__ e1x,   // stride ESTR
                                 const float* __restrict__ e2x,   // stride ESTR
                                 float* __restrict__ s1,          // stride 128, in/out
                                 float* __restrict__ s2,          // stride 128, in/out
                                 const float* __restrict__ w1,
                                 const float* __restrict__ w2) {
    long t = blockIdx.x * (long)blockDim.x + threadIdx.x;   // float4 granularity over N1 rows
    if (t >= (long)N1_C * 32) return;
    int row = (int)(t >> 5), j = (int)(t & 31);

    float4 g1, g2;
    {
        float4 x = *(const float4*)(e1x + (size_t)row * ESTR + j * 4);
        float4 s = *(const float4*)(s1 + (size_t)row * EMB_C + j * 4);
        g1.x = x.x + s.x + x.x * s.x;  g1.y = x.y + s.y + x.y * s.y;
        g1.z = x.z + s.z + x.z * s.z;  g1.w = x.w + s.w + x.w * s.w;
    }
    bool has2 = (row < N2_C);
    if (has2) {
        float4 x = *(const float4*)(e2x + (size_t)row * ESTR + j * 4);
        float4 s = *(const float4*)(s2 + (size_t)row * EMB_C + j * 4);
        g2.x = x.x + s.x + x.x * s.x;  g2.y = x.y + s.y + x.y * s.y;
        g2.z = x.z + s.z + x.z * s.z;  g2.w = x.w + s.w + x.w * s.w;
    }
    if (row < NU_C) {                    // transfer coupling on shared-user rows
        float a = w1[row], b = w2[row];
        float cx = a * g1.x + b * g2.x, cy = a * g1.y + b * g2.y;
        float cz = a * g1.z + b * g2.z, cw = a * g1.w + b * g2.w;
        g1.x = (cx + g1.x) * 0.5f; g1.y = (cy + g1.y) * 0.5f;
        g1.z = (cz + g1.z) * 0.5f; g1.w = (cw + g1.w) * 0.5f;
        g2.x = (cx + g2.x) * 0.5f; g2.y = (cy + g2.y) * 0.5f;
        g2.z = (cz + g2.z) * 0.5f; g2.w = (cw + g2.w) * 0.5f;
    }
    *(float4*)(s1 + (size_t)row * EMB_C + j * 4) = g1;
    if (has2) *(float4*)(s2 + (size_t)row * EMB_C + j * 4) = g2;
}

// ---------------- WMMA-based per-half L2 norm ----------------
// One wave normalizes 16 rows. Row sums-of-squares via V_WMMA_F32_16X16X4_F32:
//   A[m][k] = partial sum of squares of row m, K-phase k (per 32b-A VGPR layout:
//   lanes 0-15 hold K=0,1 of rows 0-15; lanes 16-31 hold K=2,3), B = ones
//   => D[m][n] = full sum of squares of row m, replicated over n.
#define TP 132   // padded LDS row stride (132 % 64 = 4 -> conflict-free K-phase reads)
__global__ void l2norm_wmma(const float* __restrict__ g,   // stride 128
                            float* __restrict__ e_out,     // stride ESTR, 128-col slice base
                            int n) {
    __shared__ float tile [4][16 * TP];
    __shared__ float sums [4][32];   // [half*16 + row]
    __shared__ float scl  [4][32];

    int wave = threadIdx.x >> 5, lane = threadIdx.x & 31;
    long r0 = ((long)blockIdx.x * 4 + wave) * 16;
    if (r0 >= n) return;                         // wave-uniform: EXEC stays full below
    float* T = tile[wave];

    // stage 16 rows (coalesced float4 per lane)
    for (int i = 0; i < 16; ++i) {
        long r = r0 + i; if (r >= n) r = n - 1;
        float4 v = *(const float4*)(g + (size_t)r * EMB_C + lane * 4);
        *(float4*)(T + i * TP + lane * 4) = v;
    }

    int mrow = lane & 15;
    int off  = (lane < 16) ? 0 : 2;
    v2f ones; ones.x = 1.f; ones.y = 1.f;

    #pragma unroll
    for (int h = 0; h < 2; ++h) {
        float p0 = 0.f, p1 = 0.f;
        #pragma unroll
        for (int j = 0; j < 16; ++j) {
            float a = T[mrow * TP + h * 64 + 4 * j + off];
            float b = T[mrow * TP + h * 64 + 4 * j + off + 1];
            p0 += a * a;
            p1 += b * b;
        }
        v2f A; A.x = p0; A.y = p1;
        v8f c = {0.f, 0.f, 0.f, 0.f, 0.f, 0.f, 0.f, 0.f};
        c = __builtin_amdgcn_wmma_f32_16x16x4_f32(
                false, A, false, ones, (short)0, c, false, false);
        // D layout: VGPR v, lanes 0-15 -> row v; lanes 16-31 -> row v+8
        if (lane == 0) {
            sums[wave][h*16+0]=c[0]; sums[wave][h*16+1]=c[1];
            sums[wave][h*16+2]=c[2]; sums[wave][h*16+3]=c[3];
            sums[wave][h*16+4]=c[4]; sums[wave][h*16+5]=c[5];
            sums[wave][h*16+6]=c[6]; sums[wave][h*16+7]=c[7];
        }
        if (lane == 16) {
            sums[wave][h*16+ 8]=c[0]; sums[wave][h*16+ 9]=c[1];
            sums[wave][h*16+10]=c[2]; sums[wave][h*16+11]=c[3];
            sums[wave][h*16+12]=c[4]; sums[wave][h*16+13]=c[5];
            sums[wave][h*16+14]=c[6]; sums[wave][h*16+15]=c[7];
        }
    }

    // scale = 1 / max(sqrt(ssq), 1e-12); lane i covers (half i/16, row i%16)
    {
        float ssq = sums[wave][lane];
        float nv  = sqrtf(ssq);
        scl[wave][lane] = 1.f / fmaxf(nv, 1e-12f);
    }

    // write normalized rows: lanes 0-15 cover half 0 (cols 0..63), 16-31 half 1
    for (int i = 0; i < 16; ++i) {
        long r = r0 + i; if (r >= n) break;
        float4 v = *(const float4*)(T + i * TP + lane * 4);
        float sc = scl[wave][((lane >= 16) ? 16 : 0) + i];
        v.x *= sc; v.y *= sc; v.z *= sc; v.w *= sc;
        *(float4*)(e_out + (size_t)r * ESTR + lane * 4) = v;
    }
}

// ---------------- gather row-permuted outputs ----------------
__global__ void gather_out(const float* __restrict__ e1,
                           const float* __restrict__ e2,
                           float* __restrict__ out) {
    const long ROWS = (long)D1U_C + D1I_C + NS_C + (NU_C - D1U_C) + D2I_C; // 270000
    long t = blockIdx.x * (long)blockDim.x + threadIdx.x;                  // float4 units
    if (t >= ROWS * 96) return;
    long row = t / 96, j = t % 96;
    const float* src;
    long b0 = D1U_C;                 // end d1_u
    long b1 = b0 + D1I_C;            // end d1_i
    long b2 = b1 + NS_C;             // end d2_u part A
    long b3 = b2 + (NU_C - D1U_C);   // end d2_u part B
    if      (row < b0) src = e1 + (size_t) row              * ESTR;
    else if (row < b1) src = e1 + (size_t)(NU_C + row - b0) * ESTR;
    else if (row < b2) src = e2 + (size_t)(row - b1)        * ESTR;
    else if (row < b3) src = e2 + (size_t)(D1U_C + row - b2)* ESTR;
    else               src = e2 + (size_t)(NU_C + row - b3) * ESTR;
    ((float4*)out)[t] = ((const float4*)src)[j];
}

// ---------------- orchestration ----------------
extern "C" void kernel_launch(void* const* d_in, const int* in_sizes, int n_in,
                              void* d_out, int out_size, void* d_ws, size_t ws_size,
                              hipStream_t stream) {
    const float* d1_user = (const float*)d_in[0];
    const float* d1_item = (const float*)d_in[1];
    const float* d2_user = (const float*)d_in[2];
    const float* d2_item = (const float*)d_in[3];
    const int*   d1_rows = (const int*)  d_in[4];
    const int*   d1_cols = (const int*)  d_in[5];
    const float* d1_vals = (const float*)d_in[6];
    const int*   d2_rows = (const int*)  d_in[7];
    const int*   d2_cols = (const int*)  d_in[8];
    const float* d2_vals = (const float*)d_in[9];
    const float* d1_deg  = (const float*)d_in[10];
    const float* d2_deg  = (const float*)d_in[11];
    float* out = (float*)d_out;

    float* e1 = (float*)d_ws;                       // N1 x 384
    float* e2 = e1 + (size_t)N1_C * ESTR;           // N2 x 384
    float* s1 = e2 + (size_t)N2_C * ESTR;           // N1 x 128
    float* s2 = s1 + (size_t)N1_C * EMB_C;          // N2 x 128

    // layer-0 features
    init_x0<<<(N1_C * 32 + 255) / 256, 256, 0, stream>>>(d1_user, d1_item, e1, NU_C, N1_C);
    init_x0<<<(N2_C * 32 + 255) / 256, 256, 0, stream>>>(d2_user, d2_item, e2, NU_C, N2_C);

    for (int l = 0; l < 2; ++l) {
        zero_f4<<<2048, 256, 0, stream>>>(s1, (long)N1_C * 32);
        zero_f4<<<2048, 256, 0, stream>>>(s2, (long)N2_C * 32);

        spmm_atomic<<<8192, 256, 0, stream>>>(d1_rows, d1_cols, d1_vals,
                                              e1 + (size_t)l * EMB_C, s1, NNZ1_C);
        spmm_atomic<<<8192, 256, 0, stream>>>(d2_rows, d2_cols, d2_vals,
                                              e2 + (size_t)l * EMB_C, s2, NNZ2_C);

        combine_transfer<<<((long)N1_C * 32 + 255) / 256, 256, 0, stream>>>(
            e1 + (size_t)l * EMB_C, e2 + (size_t)l * EMB_C, s1, s2, d1_deg, d2_deg);

        l2norm_wmma<<<(N1_C / 16 + 3) / 4, 128, 0, stream>>>(
            s1, e1 + (size_t)(l + 1) * EMB_C, N1_C);
        l2norm_wmma<<<(N2_C / 16 + 3) / 4, 128, 0, stream>>>(
            s2, e2 + (size_t)(l + 1) * EMB_C, N2_C);
    }

    const long OUT_ROWS = 270000;
    gather_out<<<(OUT_ROWS * 96 + 255) / 256, 256, 0, stream>>>(e1, e2, out);
}